// SimpleCarDynamics_62835371541161
// MI455X (gfx1250) — compile-verified
//
#include <hip/hip_runtime.h>
#include <hip/hip_bf16.h>

typedef float v2f __attribute__((ext_vector_type(2)));
typedef float v8f __attribute__((ext_vector_type(8)));

#define T_STEPS   128
#define NXCH      17
#define TRAJ_WG   16
#define NTHREADS  256
#define ELEMS     (T_STEPS * TRAJ_WG)   // 2048 elements per workgroup
#define MAP_PX    800

// ---------------------------------------------------------------------------
// Inclusive cumsum over time of a 128x16 (time x trajectory) f32 matrix in
// LDS, executed by ONE wave (EXEC must be all ones) using the matrix pipe:
//   per 16-row block i:  P_i = Ltri(16x16) . V_i   via 4 chained
//   V_WMMA_F32_16X16X4_F32 (K=4 slices), then add running carry and
//   propagate carry from row M=15 (VGPR7, lanes 16..31) with a shuffle.
// ---------------------------------------------------------------------------
__device__ __forceinline__ void wmma_scan_128x16(float* V, int lane) {
  const int n   = lane & 15;   // trajectory column; also A-matrix row M
  const int hi  = lane >> 4;   // half-wave: 0 -> K base 0, 1 -> K base 2
  const int klo = hi << 1;

  // A fragments: lower-triangular ones, L[m][c] = (c <= m).
  // 32-bit A 16x4 layout: VGPR0 = K=klo, VGPR1 = K=klo+1, M = lane&15.
  v2f a[4];
#pragma unroll
  for (int k = 0; k < 4; ++k) {
    const int c0 = 4 * k + klo;
    a[k].x = (n >= c0    ) ? 1.0f : 0.0f;
    a[k].y = (n >= c0 + 1) ? 1.0f : 0.0f;
  }

  float carry = 0.0f;
#pragma unroll
  for (int i = 0; i < 8; ++i) {
    v8f acc = {};
#pragma unroll
    for (int k = 0; k < 4; ++k) {
      const int r0 = i * 16 + 4 * k + klo;     // time row of this K slice
      v2f b;
      b.x = V[r0 * 16 + n];                    // B[K=klo  ][N=n]
      b.y = V[r0 * 16 + 16 + n];               // B[K=klo+1][N=n]
      acc = __builtin_amdgcn_wmma_f32_16x16x4_f32(
          false, a[k], false, b, (short)0, acc, false, false);
    }
#pragma unroll
    for (int j = 0; j < 8; ++j) acc[j] += carry;
    // row M=15 (= carry for next block) sits in component 7 of lanes 16..31
    carry = __shfl(acc[7], 16 + n, 32);
#pragma unroll
    for (int j = 0; j < 8; ++j) {
      const int m = j + (hi << 3);             // D layout: M = j + 8*hi
      V[(i * 16 + m) * 16 + n] = acc[j];
    }
  }
}

// ---------------------------------------------------------------------------
// One workgroup handles 16 trajectories x 128 timesteps.
// ---------------------------------------------------------------------------
__global__ __launch_bounds__(NTHREADS)
void car_dyn_kernel(const float* __restrict__ state,
                    const float* __restrict__ ctrl,
                    const float* __restrict__ Hmap,
                    const float* __restrict__ Nmap,
                    float* __restrict__ out) {
  __shared__ float sS1[ELEMS];  // yaw-increments -> yaw-delta -> yaw
  __shared__ float sS2[ELEMS];  // dx -> X-delta  -> roll
  __shared__ float sS3[ELEMS];  // dy -> Y-delta  -> pitch
  __shared__ float sVX[ELEMS];
  __shared__ float sWZ[ELEMS];
  __shared__ float sNZ[ELEMS];

  const int th   = threadIdx.x;
  const int t    = th & 127;          // consecutive lanes -> consecutive t
  const int wave = th >> 5;
  const int trajBase = blockIdx.x * TRAJ_WG;

  // ---- Phase A: per-step control math, passthrough output channels -------
#pragma unroll
  for (int i = 0; i < 8; ++i) {
    const int trajL = (th >> 7) + 2 * i;
    const int idx   = t * 16 + trajL;                   // LDS: time-major
    const int base  = (trajBase + trajL) * T_STEPS + t;
    const int sB = base * NXCH, cB = base * 2;

    const float steer = ctrl[cB];
    const float thr   = ctrl[cB + 1];
    const float curv  = tanf(steer * 0.38f) / 0.32f;
    const float vx    = thr * 5.0f;
    float wz = vx * curv;
    const float ayc = fminf(fmaxf(vx * wz, -14.0f), 14.0f);
    wz = ayc / fminf(fmaxf(vx, 1.0f), 25.0f);

    sVX[idx] = vx;
    sWZ[idx] = wz;
    sS1[idx] = 0.02f * wz;            // scan input: DT * wz

    out[sB + 6]  = vx;                // vx
    out[sB + 7]  = 0.0f;              // vy
    out[sB + 8]  = 0.0f;              // vz
    out[sB + 9]  = state[sB + 9];     // ax passthrough
    out[sB + 14] = wz;
    out[sB + 15] = steer;
    out[sB + 16] = thr;
  }
  __syncthreads();

  // ---- Phase B: WMMA scan of yaw increments (wave 0, EXEC all ones) ------
  if (wave == 0) wmma_scan_128x16(sS1, th & 31);
  __syncthreads();

  // ---- Phase C: yaw, heading, dx/dy scan inputs --------------------------
#pragma unroll
  for (int i = 0; i < 8; ++i) {
    const int trajL = (th >> 7) + 2 * i;
    const int idx   = t * 16 + trajL;
    const int base  = (trajBase + trajL) * T_STEPS + t;
    const int sB = base * NXCH;

    const float yaw = state[sB + 5] + sS1[idx];
    sS1[idx] = yaw;                   // in-place (own slot)
    const float cy = cosf(yaw), sy = sinf(yaw);
    const float dS = sVX[idx] * 0.02f;
    sS2[idx] = dS * cy;
    sS3[idx] = dS * sy;
    out[sB + 5] = yaw;
  }
  __syncthreads();

  // ---- Phase D: two independent WMMA scans in parallel waves -------------
  if      (wave == 0) wmma_scan_128x16(sS2, th & 31);
  else if (wave == 1) wmma_scan_128x16(sS3, th & 31);
  __syncthreads();

  // ---- Phase E: positions, map gather, roll/pitch, ay --------------------
#pragma unroll
  for (int i = 0; i < 8; ++i) {
    const int trajL = (th >> 7) + 2 * i;
    const int idx   = t * 16 + trajL;
    const int base  = (trajBase + trajL) * T_STEPS + t;
    const int sB = base * NXCH;

    const float x = state[sB + 0] + sS2[idx];
    const float y = state[sB + 1] + sS3[idx];

    int ix = (int)((x + 20.0f) / 0.05f);
    int iy = (int)((y + 20.0f) / 0.05f);
    ix = min(max(ix, 0), MAP_PX - 1);
    iy = min(max(iy, 0), MAP_PX - 1);
    const int mi = iy * MAP_PX + ix;
    const float z  = Hmap[mi];
    const float nx = Nmap[3 * mi + 0];
    const float ny = Nmap[3 * mi + 1];
    const float nz = Nmap[3 * mi + 2];

    const float yaw = sS1[idx];
    const float cy = cosf(yaw), sy = sinf(yaw);
    // left = n x heading, heading = (cy, sy, 0)
    const float lx = -nz * sy;
    const float ly =  nz * cy;
    const float lz =  nx * sy - ny * cy;
    // forward = left x n ; only z needed: fz = lx*ny - ly*nx
    const float fz = lx * ny - ly * nx;
    const float roll  =  asinf(lz);
    const float pitch = -asinf(fz);
    const float ay = sVX[idx] * sWZ[idx] + 9.8f * sinf(roll);

    out[sB + 0]  = x;
    out[sB + 1]  = y;
    out[sB + 2]  = z;
    out[sB + 3]  = roll;
    out[sB + 4]  = pitch;
    out[sB + 10] = ay;

    sNZ[idx] = nz;
    sS2[idx] = roll;                  // reuse scan buffers (own slot)
    sS3[idx] = pitch;
  }
  __syncthreads();

  // ---- Phase F: time-diffs wx/wy, az -------------------------------------
#pragma unroll
  for (int i = 0; i < 8; ++i) {
    const int trajL = (th >> 7) + 2 * i;
    const int idx   = t * 16 + trajL;
    const int base  = (trajBase + trajL) * T_STEPS + t;
    const int sB = base * NXCH;

    float wx, wy;
    if (t == 0) {
      wx = state[sB + 12];
      wy = state[sB + 13];
    } else {
      wx = (sS2[idx] - sS2[idx - 16]) / 0.02f;
      wy = (sS3[idx] - sS3[idx - 16]) / 0.02f;
    }
    const float az = -sVX[idx] * wy + 9.8f * sNZ[idx];
    out[sB + 11] = az;
    out[sB + 12] = wx;
    out[sB + 13] = wy;
  }
}

extern "C" void kernel_launch(void* const* d_in, const int* in_sizes, int n_in,
                              void* d_out, int out_size, void* d_ws, size_t ws_size,
                              hipStream_t stream) {
  const float* state = (const float*)d_in[0];   // [M,K,T,17] f32
  const float* ctrl  = (const float*)d_in[1];   // [M,K,T,2]  f32
  const float* Hmap  = (const float*)d_in[2];   // [800,800]  f32
  const float* Nmap  = (const float*)d_in[3];   // [800,800,3] f32
  float* out = (float*)d_out;                   // [M,K,T,17] f32

  const int nTraj = in_sizes[0] / (T_STEPS * NXCH);   // M*K = 16384
  dim3 grid(nTraj / TRAJ_WG);                          // 1024 workgroups
  dim3 block(NTHREADS);
  car_dyn_kernel<<<grid, block, 0, stream>>>(state, ctrl, Hmap, Nmap, out);
}